// FSiAM_23132693856348
// MI455X (gfx1250) — compile-verified
//
#include <hip/hip_runtime.h>
#include <hip/hip_bf16.h>

typedef float v2f __attribute__((ext_vector_type(2)));
typedef float v8f __attribute__((ext_vector_type(8)));

#define BATCH 16
#define CH    256
#define NN    4096

// ---------------------------------------------------------------------------
// Kernel 1: m[row] = mean over 4096 floats of row `row` (row = b*CH + c).
// One wave32 per row; coalesced float4 streaming; shuffle reduction.
// ---------------------------------------------------------------------------
__global__ void FSiAM_row_mean(const float* __restrict__ x,
                               float* __restrict__ m) {
    const int lane = threadIdx.x & 31;
    const int wave = threadIdx.x >> 5;
    const int row  = blockIdx.x * 8 + wave;        // 8 waves per 256-thread block
    const float* xr = x + (size_t)row * NN;

    float s = 0.0f;
#pragma unroll 4
    for (int i = 0; i < NN / 128; ++i) {           // 32 iters, 512B contiguous per wave-iter
        const float4 v = *(const float4*)(xr + i * 128 + lane * 4);
        s += v.x + v.y + v.z + v.w;
    }
#pragma unroll
    for (int off = 16; off > 0; off >>= 1)
        s += __shfl_xor(s, off, 32);

    if (lane == 0) m[row] = s * (1.0f / (float)NN);
}

// ---------------------------------------------------------------------------
// Kernel 2: scores[b,n] = sum_c x[b,c,n] * m[b,c]; out = sigmoid(scores).
// Per wave: one 16-wide n-tile; K-loop over c in chunks of 4 using
// V_WMMA_F32_16X16X4_F32 with A = m-chunk replicated across M rows and
// B = 4x16 tile of x (the bandwidth-critical operand, read exactly once).
// ---------------------------------------------------------------------------
__global__ void FSiAM_matvec_sigmoid(const float* __restrict__ x,
                                     const float* __restrict__ m,
                                     float* __restrict__ out) {
    __shared__ float sm[CH];
    const int b    = blockIdx.y;
    const int lane = threadIdx.x & 31;
    const int wave = threadIdx.x >> 5;

    // Stage this batch's m vector in LDS (blockDim.x == CH == 256).
    sm[threadIdx.x] = m[b * CH + threadIdx.x];
    __syncthreads();

    // A-matrix 16x4 f32 layout: lanes 0-15 -> M=0-15 with K={0,1},
    // lanes 16-31 -> M=0-15 with K={2,3}. B mirrors with N on lanes.
    const int koff = (lane < 16) ? 0 : 2;
    const int ncol = (blockIdx.x * 8 + wave) * 16 + (lane & 15);
    const float* xb = x + (size_t)b * CH * NN;

    v8f acc = {};
    for (int c0 = 0; c0 < CH; c0 += 4) {
        const float* xp = xb + (size_t)(c0 + koff) * NN + ncol;
        // Prefetch the next c-chunk of the B stream (global_prefetch_b8).
        __builtin_prefetch(xp + (size_t)4 * NN, 0, 1);

        v2f Av, Bv;
        Av.x = sm[c0 + koff];        // A[M][koff]   (replicated over M)
        Av.y = sm[c0 + koff + 1];    // A[M][koff+1]
        Bv.x = xp[0];                // B[koff][n]
        Bv.y = xp[NN];               // B[koff+1][n]

        // D = A x B + C   (8 args: neg_a, A, neg_b, B, c_mod, C, reuse_a, reuse_b)
        acc = __builtin_amdgcn_wmma_f32_16x16x4_f32(
            false, Av, false, Bv, (short)0, acc, false, false);
    }

    // All M rows of D are identical; D VGPR0 lane L holds scores[n-tile + (L&15)].
    const float s   = acc[0];
    const float sig = 1.0f / (1.0f + __expf(-s));
    if (lane < 16)
        out[b * NN + (blockIdx.x * 8 + wave) * 16 + lane] = sig;
}

// ---------------------------------------------------------------------------
extern "C" void kernel_launch(void* const* d_in, const int* in_sizes, int n_in,
                              void* d_out, int out_size, void* d_ws, size_t ws_size,
                              hipStream_t stream) {
    const float* x = (const float*)d_in[0];
    float* out = (float*)d_out;
    float* m   = (float*)d_ws;                 // 4096 floats = 16 KB scratch

    // Kernel 1: 4096 rows, 8 rows (waves) per block.
    FSiAM_row_mean<<<dim3((BATCH * CH) / 8), dim3(256), 0, stream>>>(x, m);

    // Kernel 2: grid.x covers n in 128-wide blocks (8 waves x 16), grid.y = batch.
    FSiAM_matvec_sigmoid<<<dim3(NN / 128, BATCH), dim3(256), 0, stream>>>(x, m, out);
}